// DGL_VGAE_70712341561937
// MI455X (gfx1250) — compile-verified
//
#include <hip/hip_runtime.h>
#include <hip/hip_bf16.h>

#define N_NODES 12288
#define FDIM    512
#define H1DIM   32
#define H2DIM   16

typedef __attribute__((ext_vector_type(2))) float v2f;
typedef __attribute__((ext_vector_type(8))) float v8f;

// ---------------------------------------------------------------- utilities
__global__ void k_zero(float* __restrict__ p, int n) {
    int i = blockIdx.x * blockDim.x + threadIdx.x;
    if (i < n) p[i] = 0.0f;
}

__global__ void k_deg(const int* __restrict__ src, const int* __restrict__ dst,
                      float* __restrict__ deg_out, float* __restrict__ deg_in, int E) {
    int e = blockIdx.x * blockDim.x + threadIdx.x;
    if (e < E) {
        atomicAdd(&deg_out[src[e]], 1.0f);
        atomicAdd(&deg_in[dst[e]], 1.0f);
    }
}

__global__ void k_norm(const float* __restrict__ deg_out, const float* __restrict__ deg_in,
                       float* __restrict__ n_src, float* __restrict__ n_dst) {
    int i = blockIdx.x * blockDim.x + threadIdx.x;
    if (i < N_NODES) {
        float d0 = deg_out[i], d1 = deg_in[i];
        n_src[i] = (d0 > 0.0f) ? rsqrtf(fmaxf(d0, 1.0f)) : 0.0f;
        n_dst[i] = (d1 > 0.0f) ? rsqrtf(fmaxf(d1, 1.0f)) : 0.0f;
    }
}

// ------------------------------------------------ layer-1 projection (WMMA)
// p[i][c] = n_src[i] * (feat[i,:] @ W1[:,c])   (exact reorder of reference)
// One wave per 16-row tile, two 16-col C tiles (H1=32), K=512 in steps of 4.
__global__ void k_gemm1(const float* __restrict__ feat, const float* __restrict__ W1,
                        const float* __restrict__ n_src, float* __restrict__ p) {
    int wave  = threadIdx.x >> 5;
    int lane  = threadIdx.x & 31;
    int itile = blockIdx.x * 8 + wave;
    int ibase = itile * 16;
    int m     = lane & 15;
    int koff  = (lane < 16) ? 0 : 2;

    const float* arow = feat + (size_t)(ibase + m) * FDIM + koff;
    v8f c0 = {}; v8f c1 = {};
    for (int kk = 0; kk < FDIM; kk += 4) {
        v2f a = *(const v2f*)(arow + kk);
        int kb = kk + koff;
        v2f b0, b1;
        b0.x = W1[(size_t)kb * H1DIM + m];            // B[k][n], n = m (cols 0..15)
        b0.y = W1[(size_t)(kb + 1) * H1DIM + m];
        b1.x = W1[(size_t)kb * H1DIM + 16 + m];       // cols 16..31
        b1.y = W1[(size_t)(kb + 1) * H1DIM + 16 + m];
        c0 = __builtin_amdgcn_wmma_f32_16x16x4_f32(false, a, false, b0, (short)0, c0, false, false);
        c1 = __builtin_amdgcn_wmma_f32_16x16x4_f32(false, a, false, b1, (short)0, c1, false, false);
    }
    int rbase = (lane < 16) ? 0 : 8;
    #pragma unroll
    for (int r = 0; r < 8; ++r) {
        int row = ibase + r + rbase;
        float s = n_src[row];
        p[(size_t)row * H1DIM + m]      = s * c0[r];
        p[(size_t)row * H1DIM + 16 + m] = s * c1[r];
    }
}

// --------------------------------------------------- edge scatter-add SpMM
__global__ void k_scatter(const int* __restrict__ src, const int* __restrict__ dst,
                          const float* __restrict__ p, float* __restrict__ agg, int E) {
    int tid = blockIdx.x * blockDim.x + threadIdx.x;
    if (tid < E * H1DIM) {
        int e = tid >> 5;
        int c = tid & 31;
        float v = p[(size_t)src[e] * H1DIM + c];
        atomicAdd(&agg[(size_t)dst[e] * H1DIM + c], v);
    }
}

// h1 = relu(n_dst*agg + b1); overwrite p with n_src*h1 (input for layer 2 agg)
__global__ void k_h1(const float* __restrict__ agg, const float* __restrict__ n_src,
                     const float* __restrict__ n_dst, const float* __restrict__ b1,
                     float* __restrict__ p) {
    int i = blockIdx.x * blockDim.x + threadIdx.x;
    if (i < N_NODES * H1DIM) {
        int node = i >> 5;
        int c    = i & 31;
        float h = fmaxf(n_dst[node] * agg[i] + b1[c], 0.0f);
        p[i] = n_src[node] * h;
    }
}

// ------------------------------------ head: mu / logvar (tiny [N,32]@[32,16])
__global__ void k_head(const float* __restrict__ agg, const float* __restrict__ n_dst,
                       const float* __restrict__ W2, const float* __restrict__ b2,
                       const float* __restrict__ W3, const float* __restrict__ b3,
                       float* __restrict__ out) {
    int t = blockIdx.x * blockDim.x + threadIdx.x;
    if (t < N_NODES * H2DIM) {
        int i = t >> 4;
        int j = t & 15;
        float nd = n_dst[i];
        float mu = b2[j], lv = b3[j];
        #pragma unroll
        for (int c = 0; c < H1DIM; ++c) {
            float h = nd * agg[(size_t)i * H1DIM + c];
            mu += h * W2[c * H2DIM + j];
            lv += h * W3[c * H2DIM + j];
        }
        size_t NN = (size_t)N_NODES * N_NODES;
        out[NN + (size_t)i * H2DIM + j] = mu;
        out[NN + (size_t)N_NODES * H2DIM + (size_t)i * H2DIM + j] = lv;
    }
}

// ------------------------------- adj_rec = z @ z.T  (store-bandwidth bound)
// One wave per 16x16 output tile; K=16 -> 4 f32 WMMAs; NT stores (604 MB stream).
__global__ void k_zzt(const float* __restrict__ z, float* __restrict__ out) {
    int wave  = threadIdx.x >> 5;
    int lane  = threadIdx.x & 31;
    int jtile = blockIdx.x * 8 + wave;
    int itile = blockIdx.y;
    int m     = lane & 15;
    int koff  = (lane < 16) ? 0 : 2;

    const float* za = z + (size_t)(itile * 16 + m) * H2DIM + koff;
    const float* zb = z + (size_t)(jtile * 16 + m) * H2DIM + koff;

    v8f c = {};
    #pragma unroll
    for (int kk = 0; kk < H2DIM; kk += 4) {
        v2f a = *(const v2f*)(za + kk);
        v2f b = *(const v2f*)(zb + kk);
        c = __builtin_amdgcn_wmma_f32_16x16x4_f32(false, a, false, b, (short)0, c, false, false);
    }

    int rbase = (lane < 16) ? 0 : 8;
    size_t obase = (size_t)(itile * 16) * N_NODES + (size_t)jtile * 16 + m;
    #pragma unroll
    for (int r = 0; r < 8; ++r) {
        __builtin_nontemporal_store(c[r], out + obase + (size_t)(r + rbase) * N_NODES);
    }
}

// --------------------------------------------------------------- dispatcher
extern "C" void kernel_launch(void* const* d_in, const int* in_sizes, int n_in,
                              void* d_out, int out_size, void* d_ws, size_t ws_size,
                              hipStream_t stream) {
    const float* feat = (const float*)d_in[0];
    const int*   src  = (const int*)d_in[1];
    const int*   dst  = (const int*)d_in[2];
    const float* W1   = (const float*)d_in[3];
    const float* b1   = (const float*)d_in[4];
    const float* W2   = (const float*)d_in[5];
    const float* b2   = (const float*)d_in[6];
    const float* W3   = (const float*)d_in[7];
    const float* b3   = (const float*)d_in[8];
    float*       out  = (float*)d_out;

    const int E = in_sizes[1];
    const int N = N_NODES;

    float* w       = (float*)d_ws;
    float* deg_out = w;                 // [N]
    float* deg_in  = w + N;             // [N]
    float* n_src   = w + 2 * N;         // [N]
    float* n_dst   = w + 3 * N;         // [N]
    float* p       = w + 4 * N;         // [N,32]
    float* agg     = w + 4 * N + 32 * N;// [N,32]

    const int B = 256;

    // degrees + norms
    k_zero<<<(2 * N + B - 1) / B, B, 0, stream>>>(deg_out, 2 * N);
    k_deg<<<(E + B - 1) / B, B, 0, stream>>>(src, dst, deg_out, deg_in, E);
    k_norm<<<(N + B - 1) / B, B, 0, stream>>>(deg_out, deg_in, n_src, n_dst);

    // layer 1: project-first GEMM (WMMA f32), scatter, relu+rescale
    k_gemm1<<<(N / 16) / 8, B, 0, stream>>>(feat, W1, n_src, p);
    k_zero<<<(32 * N + B - 1) / B, B, 0, stream>>>(agg, 32 * N);
    k_scatter<<<(E * H1DIM + B - 1) / B, B, 0, stream>>>(src, dst, p, agg, E);
    k_h1<<<(32 * N + B - 1) / B, B, 0, stream>>>(agg, n_src, n_dst, b1, p);

    // layer 2 aggregation (shared by mu & logvar), then head
    k_zero<<<(32 * N + B - 1) / B, B, 0, stream>>>(agg, 32 * N);
    k_scatter<<<(E * H1DIM + B - 1) / B, B, 0, stream>>>(src, dst, p, agg, E);
    k_head<<<(N * H2DIM + B - 1) / B, B, 0, stream>>>(agg, n_dst, W2, b2, W3, b3, out);

    // adj_rec = mu @ mu.T   (read mu back from d_out tail; stream-ordered)
    const float* zmu = out + (size_t)N * N;
    dim3 grid((N / 16) / 8, N / 16);
    k_zzt<<<grid, B, 0, stream>>>(zmu, out);
}